// DetBenchPredict_65120294142003
// MI455X (gfx1250) — compile-verified
//
#include <hip/hip_runtime.h>
#include <stdint.h>

// ---------------- problem constants ----------------
#define NUM_CLASSES     90
#define NUM_ANCHORS     110484
#define BATCH           8
#define TOTAL           (NUM_ANCHORS * NUM_CLASSES)   // 9,943,560 per image
#define MAX_DET_POINTS  5000
#define MAX_DET         100
#define IOU_T           0.5f

// ---------------- tuning ----------------
#define CHUNK      8192                          // elements per TDM tile (32 KB)
#define NCHUNK     ((TOTAL + CHUNK - 1) / CHUNK) // 1214
#define SCAN_BLOCKS 256                          // blocks per image for scan passes
#define HIST_BINS  16384
#define BIN_SHIFT  18                            // 32 - 14 bits
#define CAND_MAX   8192                          // sorted-candidate capacity per image

typedef __attribute__((ext_vector_type(4))) unsigned int u32x4;
typedef __attribute__((ext_vector_type(4))) int          i32x4;
typedef __attribute__((ext_vector_type(8))) int          i32x8;
typedef __attribute__((address_space(3)))   char         lds_char_t;

// monotone mapping float -> u32 (bigger float => bigger uint)
__device__ __forceinline__ unsigned sortable_bits(float x) {
  unsigned u = __float_as_uint(x);
  return (u & 0x80000000u) ? ~u : (u | 0x80000000u);
}
__device__ __forceinline__ float unsortable_bits(unsigned s) {
  unsigned u = (s & 0x80000000u) ? (s & 0x7FFFFFFFu) : ~s;
  return __uint_as_float(u);
}

// ---------------- CDNA5 Tensor Data Mover: issue 1-D global -> LDS DMA ----------
// Builds a D# per CDNA5 ISA (group0: count/lds/global/type,
// group1: data_size + dims/strides) and issues TENSOR_LOAD_TO_LDS (TENSORcnt++).
// Caller controls waiting, enabling a 2-deep double-buffered DMA pipeline.
__device__ __forceinline__ void tdm_issue_1d(const float* gsrc, void* lds_dst, int elems) {
  unsigned lds_off = (unsigned)(unsigned long long)(lds_char_t*)lds_dst;
  unsigned long long ga = (unsigned long long)gsrc;
  u32x4 g0 = {};
  g0[0] = 1u;                                           // count = 1 valid descriptor
  g0[1] = lds_off;                                      // lds_addr
  g0[2] = (unsigned)ga;                                 // global_addr[31:0]
  g0[3] = (unsigned)((ga >> 32) & 0x1FFFFFFull)         // global_addr[56:32]
        | (2u << 30);                                   // type = 2 ("image")
  i32x8 g1 = {};
  g1[0] = 0x00020000;                                   // data_size = 2 -> 4 bytes
  g1[1] = (int)(((unsigned)elems & 0xFFFFu) << 16);     // tensor_dim0[15:0]
  g1[2] = (int)((((unsigned)elems >> 16) & 0xFFFFu)     // tensor_dim0[31:16]
        | (1u << 16));                                  // tensor_dim1 = 1
  g1[3] = (int)(((unsigned)elems & 0xFFFFu) << 16);     // tile_dim0 = elems
  g1[5] = elems;                                        // tensor_dim0_stride (lo 32)
  i32x4 z4 = {};
#if __clang_major__ >= 23
  i32x8 z8 = {};
  __builtin_amdgcn_tensor_load_to_lds(g0, g1, z4, z4, z8, 0);
#else
  __builtin_amdgcn_tensor_load_to_lds(g0, g1, z4, z4, 0);
#endif
}

// ---------------- K0: zero the stateful scratch (histogram + counters) ----------
__global__ void k_zero(unsigned* __restrict__ hist, unsigned* __restrict__ cnt) {
  int i = blockIdx.x * blockDim.x + threadIdx.x;
  if (i < BATCH * HIST_BINS) hist[i] = 0u;
  if (i < BATCH) cnt[i] = 0u;
}

// ---------------- K1: double-buffered TDM streaming histogram ------------------
// Each block walks chunks c = blockIdx.x, +gridDim.x, ... ; wave 0 keeps up to two
// tensor DMAs in flight (issue next tile, s_wait_tensorcnt 1 for current tile).
__global__ void __launch_bounds__(256) k_hist(const float* __restrict__ cls_out,
                                              unsigned* __restrict__ g_hist) {
  __shared__ float    tile[2][CHUNK];     // 2 x 32 KB, filled by TDM
  __shared__ unsigned lhist[HIST_BINS];   // 64 KB (CDNA5: 320 KB LDS per WGP)
  const int b = blockIdx.y;
  const float* src = cls_out + (long long)b * TOTAL;

  for (int i = threadIdx.x; i < HIST_BINS; i += blockDim.x) lhist[i] = 0u;

  int c = blockIdx.x;
  bool valid = (c < NCHUNK);
  if (valid && threadIdx.x < 32) {
    tdm_issue_1d(src + c * CHUNK, tile[0], min(CHUNK, TOTAL - c * CHUNK));
  }
  int p = 0;
  while (valid) {
    const int cn = c + gridDim.x;
    const bool has_next = (cn < NCHUNK);
    if (threadIdx.x < 32) {
      if (has_next) {
        tdm_issue_1d(src + cn * CHUNK, tile[p ^ 1], min(CHUNK, TOTAL - cn * CHUNK));
        __builtin_amdgcn_s_wait_tensorcnt(1);   // current tile has landed
      } else {
        __builtin_amdgcn_s_wait_tensorcnt(0);
      }
    }
    __syncthreads();
    const int elems = min(CHUNK, TOTAL - c * CHUNK);
    for (int i = threadIdx.x; i < elems; i += blockDim.x) {
      atomicAdd(&lhist[sortable_bits(tile[p][i]) >> BIN_SHIFT], 1u);
    }
    __syncthreads();   // nobody reads tile[p] after this; safe to overwrite next iter
    c = cn; valid = has_next; p ^= 1;
  }
  __syncthreads();

  unsigned* gh = g_hist + b * HIST_BINS;
  for (int i = threadIdx.x; i < HIST_BINS; i += blockDim.x) {
    unsigned v = lhist[i];
    if (v) atomicAdd(&gh[i], v);
  }
}

// ---------------- K2: find per-image threshold bin (suffix count >= 5000) ------
__global__ void k_thresh(const unsigned* __restrict__ hist,
                         unsigned* __restrict__ thr) {
  int b = threadIdx.x;
  if (b >= BATCH) return;
  const unsigned* h = hist + b * HIST_BINS;
  unsigned acc = 0; int t = 0;
  for (int i = HIST_BINS - 1; i >= 0; --i) {
    acc += h[i];
    if (acc >= MAX_DET_POINTS) { t = i; break; }
  }
  thr[b] = (unsigned)t;
}

// ---------------- K3: compact candidates above threshold (TDM-staged) ----------
__global__ void __launch_bounds__(256) k_compact(const float* __restrict__ cls_out,
                                                 const unsigned* __restrict__ thr,
                                                 unsigned* __restrict__ cnt,
                                                 unsigned long long* __restrict__ cand) {
  __shared__ float tile[2][CHUNK];        // 2 x 32 KB, filled by TDM
  const int b = blockIdx.y;
  const float* src = cls_out + (long long)b * TOTAL;
  const unsigned t = thr[b];
  unsigned long long* cb = cand + (size_t)b * CAND_MAX;

  int c = blockIdx.x;
  bool valid = (c < NCHUNK);
  if (valid && threadIdx.x < 32) {
    tdm_issue_1d(src + c * CHUNK, tile[0], min(CHUNK, TOTAL - c * CHUNK));
  }
  int p = 0;
  while (valid) {
    const int cn = c + gridDim.x;
    const bool has_next = (cn < NCHUNK);
    if (threadIdx.x < 32) {
      if (has_next) {
        tdm_issue_1d(src + cn * CHUNK, tile[p ^ 1], min(CHUNK, TOTAL - cn * CHUNK));
        __builtin_amdgcn_s_wait_tensorcnt(1);
      } else {
        __builtin_amdgcn_s_wait_tensorcnt(0);
      }
    }
    __syncthreads();
    const int start = c * CHUNK;
    const int elems = min(CHUNK, TOTAL - start);
    for (int i = threadIdx.x; i < elems; i += blockDim.x) {
      unsigned s = sortable_bits(tile[p][i]);
      if ((s >> BIN_SHIFT) >= t) {
        int idx = start + i;                     // < 2^24, fits in 32 bits
        unsigned pos = atomicAdd(&cnt[b], 1u);
        if (pos < CAND_MAX) {
          // key: score bits (desc) then ~idx (=> smaller idx first on ties)
          cb[pos] = ((unsigned long long)s << 32) | (unsigned)(~(unsigned)idx);
        }
      }
    }
    __syncthreads();
    c = cn; valid = has_next; p ^= 1;
  }
}

// ---------------- K4: in-LDS bitonic sort + top-5000 decode/sigmoid ------------
__global__ void __launch_bounds__(1024) k_sort_decode(
    const unsigned long long* __restrict__ cand,
    const unsigned* __restrict__ cnt,
    const float* __restrict__ box_out,
    const float* __restrict__ anchors,
    float* __restrict__ dbox, float* __restrict__ dscore, int* __restrict__ dcls) {
  __shared__ unsigned long long key[CAND_MAX];   // 64 KB
  const int b = blockIdx.x;
  const int n = min((int)cnt[b], CAND_MAX);

  for (int i = threadIdx.x; i < CAND_MAX; i += blockDim.x)
    key[i] = (i < n) ? cand[(size_t)b * CAND_MAX + i] : 0ull;
  __syncthreads();

  for (unsigned k = 2; k <= CAND_MAX; k <<= 1) {
    for (unsigned j = k >> 1; j > 0; j >>= 1) {
      for (unsigned i = threadIdx.x; i < CAND_MAX; i += blockDim.x) {
        unsigned ixj = i ^ j;
        if (ixj > i) {
          bool desc = ((i & k) == 0);
          unsigned long long a = key[i], c = key[ixj];
          if (desc ? (a < c) : (a > c)) { key[i] = c; key[ixj] = a; }
        }
      }
      __syncthreads();
    }
  }

  const float* bo = box_out + (size_t)b * NUM_ANCHORS * 4;
  for (int i = threadIdx.x; i < MAX_DET_POINTS; i += blockDim.x) {
    unsigned long long kv = key[i];
    unsigned s   = (unsigned)(kv >> 32);
    unsigned idx = ~(unsigned)(kv & 0xFFFFFFFFull);
    int anchor = (int)(idx / NUM_CLASSES);
    int cls    = (int)(idx - (unsigned)anchor * NUM_CLASSES);
    anchor = min(anchor, NUM_ANCHORS - 1);

    float a0 = anchors[anchor * 4 + 0], a1 = anchors[anchor * 4 + 1];
    float a2 = anchors[anchor * 4 + 2], a3 = anchors[anchor * 4 + 3];
    float ty = bo[anchor * 4 + 0], tx = bo[anchor * 4 + 1];
    float th = bo[anchor * 4 + 2], tw = bo[anchor * 4 + 3];
    float yca = 0.5f * (a0 + a2), xca = 0.5f * (a1 + a3);
    float ha = a2 - a0, wa = a3 - a1;
    float w = expf(tw) * wa, h = expf(th) * ha;
    float yc = ty * ha + yca, xc = tx * wa + xca;

    float* d = dbox + ((size_t)b * MAX_DET_POINTS + i) * 4;
    d[0] = xc - 0.5f * w; d[1] = yc - 0.5f * h;
    d[2] = xc + 0.5f * w; d[3] = yc + 0.5f * h;
    float logit = unsortable_bits(s);
    dscore[b * MAX_DET_POINTS + i] = 1.0f / (1.0f + expf(-logit));
    dcls[b * MAX_DET_POINTS + i]   = cls;
  }
}

// ---------------- K5: class-aware greedy NMS (== reference offset trick) -------
__global__ void __launch_bounds__(128) k_nms(const float* __restrict__ dbox,
                                             const float* __restrict__ dscore,
                                             const int* __restrict__ dcls,
                                             const float* __restrict__ img_scale,
                                             float* __restrict__ out) {
  __shared__ float kb[MAX_DET][4];
  __shared__ int   kc[MAX_DET];
  __shared__ int   ksrc[MAX_DET];
  __shared__ int   nkept;
  __shared__ int   sflag;
  const int b = blockIdx.x;
  const float* boxes  = dbox   + (size_t)b * MAX_DET_POINTS * 4;
  const float* scores = dscore + b * MAX_DET_POINTS;
  const int*   clss   = dcls   + b * MAX_DET_POINTS;

  if (threadIdx.x == 0) nkept = 0;
  __syncthreads();

  for (int i = 0; i < MAX_DET_POINTS; ++i) {
    int nk = nkept;
    if (nk >= MAX_DET) break;          // uniform: only first 100 kept matter
    float x1 = boxes[i * 4 + 0], y1 = boxes[i * 4 + 1];
    float x2 = boxes[i * 4 + 2], y2 = boxes[i * 4 + 3];
    int c = clss[i];
    if (threadIdx.x == 0) sflag = 0;
    __syncthreads();

    float area = (x2 - x1) * (y2 - y1);
    int sup = 0;
    for (int j = threadIdx.x; j < nk; j += blockDim.x) {
      if (kc[j] != c) continue;        // class-aware (== big-offset trick)
      float ix = fminf(x2, kb[j][2]) - fmaxf(x1, kb[j][0]);
      float iy = fminf(y2, kb[j][3]) - fmaxf(y1, kb[j][1]);
      ix = fmaxf(ix, 0.0f); iy = fmaxf(iy, 0.0f);
      float inter = ix * iy;
      float areaj = (kb[j][2] - kb[j][0]) * (kb[j][3] - kb[j][1]);
      float uni = area + areaj - inter;
      if (inter > 0.0f && inter > IOU_T * uni) sup = 1;
    }
    if (sup) atomicOr(&sflag, 1);
    __syncthreads();
    if (threadIdx.x == 0 && sflag == 0) {
      int k = nkept;
      kb[k][0] = x1; kb[k][1] = y1; kb[k][2] = x2; kb[k][3] = y2;
      kc[k] = c; ksrc[k] = i;
      nkept = k + 1;
    }
    __syncthreads();
  }

  const float scale = img_scale[b];
  const int nk = nkept;
  for (int r = threadIdx.x; r < MAX_DET; r += blockDim.x) {
    float* o = out + ((size_t)b * MAX_DET + r) * 6;
    if (r < nk) {
      o[0] = kb[r][0] * scale; o[1] = kb[r][1] * scale;
      o[2] = kb[r][2] * scale; o[3] = kb[r][3] * scale;
      o[4] = scores[ksrc[r]];
      o[5] = (float)(kc[r] + 1);
    } else {
      o[0] = 0.0f; o[1] = 0.0f; o[2] = 0.0f; o[3] = 0.0f;
      o[4] = 0.0f; o[5] = -1.0f;
    }
  }
}

// ---------------- host side ----------------
extern "C" void kernel_launch(void* const* d_in, const int* in_sizes, int n_in,
                              void* d_out, int out_size, void* d_ws, size_t ws_size,
                              hipStream_t stream) {
  (void)in_sizes; (void)n_in; (void)out_size; (void)ws_size;
  const float* cls_out   = (const float*)d_in[0];  // (B, A, 90)
  const float* box_out   = (const float*)d_in[1];  // (B, A, 4)
  const float* anchors   = (const float*)d_in[2];  // (A, 4)
  const float* img_scale = (const float*)d_in[3];  // (B,)
  float* out = (float*)d_out;                      // (B, 100, 6)

  char* ws = (char*)d_ws;
  size_t off = 0;
  auto take = [&](size_t bytes) -> void* {
    void* p = ws + off;
    off = (off + bytes + 255) & ~(size_t)255;
    return p;
  };
  unsigned*           hist   = (unsigned*)take((size_t)BATCH * HIST_BINS * 4); // 512 KB
  unsigned*           cnt    = (unsigned*)take((size_t)BATCH * 4);
  unsigned*           thr    = (unsigned*)take((size_t)BATCH * 4);
  unsigned long long* cand   = (unsigned long long*)take((size_t)BATCH * CAND_MAX * 8); // 512 KB
  float*              dbox   = (float*)take((size_t)BATCH * MAX_DET_POINTS * 4 * 4);
  float*              dscore = (float*)take((size_t)BATCH * MAX_DET_POINTS * 4);
  int*                dcls   = (int*)take((size_t)BATCH * MAX_DET_POINTS * 4);

  k_zero<<<(BATCH * HIST_BINS + 255) / 256, 256, 0, stream>>>(hist, cnt);

  dim3 gscan(SCAN_BLOCKS, BATCH);   // each block pipelines ~5 chunks, 2-deep TDM
  k_hist<<<gscan, 256, 0, stream>>>(cls_out, hist);
  k_thresh<<<1, 32, 0, stream>>>(hist, thr);
  k_compact<<<gscan, 256, 0, stream>>>(cls_out, thr, cnt, cand);
  k_sort_decode<<<BATCH, 1024, 0, stream>>>(cand, cnt, box_out, anchors,
                                            dbox, dscore, dcls);
  k_nms<<<BATCH, 128, 0, stream>>>(dbox, dscore, dcls, img_scale, out);
}